// GCN_44624710205523
// MI455X (gfx1250) — compile-verified
//
#include <hip/hip_runtime.h>
#include <hip/hip_bf16.h>

#define N_NODES 100000
#define N_EDGES 3200000
#define FEAT    128

typedef __attribute__((ext_vector_type(2))) float v2f;
typedef __attribute__((ext_vector_type(8))) float v8f;

// ---------------- elementwise / graph kernels ----------------

__global__ void zero_f32(float* __restrict__ p, int n) {
    int i = blockIdx.x * blockDim.x + threadIdx.x;
    if (i < n) p[i] = 0.0f;
}

__global__ void degrees_kernel(const int* __restrict__ src, const int* __restrict__ dst,
                               float* __restrict__ deg_out, float* __restrict__ deg_in, int e) {
    int i = blockIdx.x * blockDim.x + threadIdx.x;
    if (i < e) {
        atomicAdd(deg_out + src[i], 1.0f);
        atomicAdd(deg_in  + dst[i], 1.0f);
    }
}

__global__ void norms_kernel(float* __restrict__ d, int n) {
    int i = blockIdx.x * blockDim.x + threadIdx.x;
    if (i < n) d[i] = rsqrtf(fmaxf(d[i], 1.0f));
}

// y[row][:] = x[row][:] * s[row], 128 cols, one float4 per thread
__global__ void scale_rows(const float* __restrict__ x, const float* __restrict__ s,
                           float* __restrict__ y, int nrows) {
    int i = blockIdx.x * blockDim.x + threadIdx.x;   // float4 index
    int total = nrows * (FEAT / 4);
    if (i < total) {
        int row = i / (FEAT / 4);
        float4 v = ((const float4*)x)[i];
        float sc = s[row];
        v.x *= sc; v.y *= sc; v.z *= sc; v.w *= sc;
        ((float4*)y)[i] = v;
    }
}

// agg[dst] += xs[src] : one wave (32 lanes) per edge, float4 per lane = 512B row
__global__ void edge_aggregate(const float* __restrict__ xs, const int* __restrict__ src,
                               const int* __restrict__ dst, float* __restrict__ agg, int e) {
    int gid  = blockIdx.x * blockDim.x + threadIdx.x;
    int edge = gid >> 5;
    int lane = threadIdx.x & 31;
    if (edge < e) {
        // src/dst are wave-uniform: force them scalar so row bases live in SGPRs
        int s = __builtin_amdgcn_readfirstlane(src[edge]);
        int d = __builtin_amdgcn_readfirstlane(dst[edge]);
        float4 v = ((const float4*)(xs + (size_t)s * FEAT))[lane];
        float* o = agg + (size_t)d * FEAT + lane * 4;
        atomicAdd(o + 0, v.x);
        atomicAdd(o + 1, v.y);
        atomicAdd(o + 2, v.z);
        atomicAdd(o + 3, v.w);
    }
}

// ---------------- fused WMMA GEMM ----------------
// C[16 x F tile] = act( (pre ? diag(pre)*A : A) @ W + bias ) * (post ? diag(post) : I)
// A: [N,128] f32, W: [128,F] row-major, bias: [F]
// block = 128 threads = 4 wave32; grid.x = N/16 row tiles.
// ACT: 0 = ReLU, 1 = sigmoid.  F is compile-time so all B/C offsets are immediates.
template <int ACT, int F>
__global__ void gemm_fused(const float* __restrict__ A, const float* __restrict__ W,
                           const float* __restrict__ bias,
                           const float* __restrict__ pre,
                           const float* __restrict__ post,
                           float* __restrict__ C) {
    __shared__ float As[16][FEAT + 4];   // 528B row stride: conflict-free, 8B-aligned pairs
    const int row0 = blockIdx.x * 16;
    const int t    = threadIdx.x;

    // cooperative A-tile load (coalesced), fused pre-scale by in_norm
    {
        int r  = t >> 3;           // 0..15
        int c0 = (t & 7) * 16;     // 16-float chunk
        float sc = pre ? pre[row0 + r] : 1.0f;
        const float* ap = A + (size_t)(row0 + r) * FEAT + c0;
        #pragma unroll
        for (int j = 0; j < 16; j += 4) {
            float4 v = *(const float4*)(ap + j);
            As[r][c0 + j + 0] = v.x * sc;
            As[r][c0 + j + 1] = v.y * sc;
            As[r][c0 + j + 2] = v.z * sc;
            As[r][c0 + j + 3] = v.w * sc;
        }
    }
    __syncthreads();

    const int wave = t >> 5;
    const int lane = t & 31;
    const int half = lane >> 4;    // 0: K/M group 0, 1: K+2 / M+8 group
    const int lid  = lane & 15;    // M index for A, N index for B/C/D

    // single base per lane for A fragments (LDS) — k0 offsets are immediates
    const float* alds = &As[lid][2 * half];

    // each wave walks column tiles with stride 4 (F=128 -> 2 tiles, F=64 -> 1)
    #pragma unroll
    for (int ct0 = 0; ct0 < (F >> 4); ct0 += 4) {
        const int ct  = ct0 + wave;
        const int col = ct << 4;
        v8f acc;
        {
            float bv = bias[col + lid];   // bias depends only on column
            #pragma unroll
            for (int i = 0; i < 8; ++i) acc[i] = bv;
        }
        // single base per lane for B fragments — k0*F offsets are immediates
        const float* wp = W + col + lid + 2 * half * F;
        #pragma unroll
        for (int k0 = 0; k0 < FEAT; k0 += 4) {
            // f32 A fragment 16x4: VGPR0 = K = 2*half, VGPR1 = K = 2*half+1 (adjacent -> b64 LDS load)
            v2f a = *(const v2f*)(alds + k0);
            v2f b;
            b.x = wp[k0 * F];        // K = k0 + 2*half
            b.y = wp[k0 * F + F];    // K = k0 + 2*half + 1
            acc = __builtin_amdgcn_wmma_f32_16x16x4_f32(
                false, a, false, b, (short)0, acc, false, false);
        }
        // epilogue: activation + optional post row-scale (out_norm), store
        const int mbase = half * 8;
        #pragma unroll
        for (int i = 0; i < 8; ++i) {
            int m = mbase + i;
            float z = acc[i];
            if (ACT == 0) z = fmaxf(z, 0.0f);
            else          z = 1.0f / (1.0f + __expf(-z));
            if (post) z *= post[row0 + m];
            C[(size_t)(row0 + m) * F + col + lid] = z;
        }
    }
}

// ---------------- launcher ----------------

extern "C" void kernel_launch(void* const* d_in, const int* in_sizes, int n_in,
                              void* d_out, int out_size, void* d_ws, size_t ws_size,
                              hipStream_t stream) {
    const float* x   = (const float*)d_in[0];
    const int*   src = (const int*)d_in[1];
    const int*   dst = (const int*)d_in[2];
    const float* W1  = (const float*)d_in[3];
    const float* b1  = (const float*)d_in[4];
    const float* W2  = (const float*)d_in[5];
    const float* b2  = (const float*)d_in[6];
    const float* Wm1 = (const float*)d_in[7];
    const float* bm1 = (const float*)d_in[8];
    const float* Wm2 = (const float*)d_in[9];
    const float* bm2 = (const float*)d_in[10];
    float* out = (float*)d_out;

    float* ws       = (float*)d_ws;
    float* out_norm = ws;                                   // N
    float* in_norm  = ws + N_NODES;                         // N
    float* bufB     = ws + 2 * (size_t)N_NODES;             // N*128
    float* bufC     = bufB + (size_t)N_NODES * FEAT;        // N*128

    const int TB   = 256;
    const int nf   = N_NODES * FEAT;
    const int rowT = N_NODES / 16;   // 6250, exact

    // degrees -> norms (both arrays contiguous in ws)
    zero_f32<<<(2 * N_NODES + TB - 1) / TB, TB, 0, stream>>>(out_norm, 2 * N_NODES);
    degrees_kernel<<<(N_EDGES + TB - 1) / TB, TB, 0, stream>>>(src, dst, out_norm, in_norm, N_EDGES);
    norms_kernel<<<(2 * N_NODES + TB - 1) / TB, TB, 0, stream>>>(out_norm, 2 * N_NODES);

    // conv1: xs = x * out_norm -> bufB ; agg -> bufC ; gemm -> bufB (post-scaled by out_norm for conv2)
    scale_rows<<<(N_NODES * (FEAT / 4) + TB - 1) / TB, TB, 0, stream>>>(x, out_norm, bufB, N_NODES);
    zero_f32<<<(nf + TB - 1) / TB, TB, 0, stream>>>(bufC, nf);
    edge_aggregate<<<(int)(((size_t)N_EDGES * 32 + TB - 1) / TB), TB, 0, stream>>>(bufB, src, dst, bufC, N_EDGES);
    gemm_fused<0, 128><<<rowT, 128, 0, stream>>>(bufC, W1, b1, in_norm, out_norm, bufB);

    // conv2: agg(bufB) -> bufC ; gemm -> bufB (no post-scale)
    zero_f32<<<(nf + TB - 1) / TB, TB, 0, stream>>>(bufC, nf);
    edge_aggregate<<<(int)(((size_t)N_EDGES * 32 + TB - 1) / TB), TB, 0, stream>>>(bufB, src, dst, bufC, N_EDGES);
    gemm_fused<0, 128><<<rowT, 128, 0, stream>>>(bufC, W2, b2, in_norm, nullptr, bufB);

    // MLP head
    gemm_fused<0, 128><<<rowT, 128, 0, stream>>>(bufB, Wm1, bm1, nullptr, nullptr, bufC);
    gemm_fused<1, 64><<<rowT, 128, 0, stream>>>(bufC, Wm2, bm2, nullptr, nullptr, out);
}